// SelfAttention_60086592471086
// MI455X (gfx1250) — compile-verified
//
#include <hip/hip_runtime.h>
#include <hip/hip_bf16.h>
#include <math.h>

// Problem constants (reference: N=65536, D=512, fp32 everywhere).
//
// Roofline: after reassociating x = v @ s into
//   x_i = inputs_i · (value_w.T @ s) + value_b·s,
// the whole op is two streaming passes over inputs (2 x 128 MB) plus ~5 MB of
// side traffic => ~261 MB / 23.3 TB/s ~= 11 us. Compute is ~134 MFLOP fp32,
// so the V_WMMA_F32_16X16X4_F32 matvec mapping in pass B (15/16 of columns
// wasted) is still far below the memory time — the waste is free, and f32
// WMMA keeps full precision.
#define N_ROWS 65536
#define DIM    512
#define GRID_A 1024               // pass-A blocks
#define ROWS_A (N_ROWS / GRID_A)  // 64 rows / block (8 per wave)
#define GRID_B 512                // pass-B blocks
#define ROWS_B (N_ROWS / GRID_B)  // 128 rows / block (16 per wave, 8 waves)

typedef __attribute__((ext_vector_type(2))) float v2f;
typedef __attribute__((ext_vector_type(8))) float v8f;

// ---------------------------------------------------------------------------
// Pass A: one sweep over inputs (128 MB). Wave-per-row, lanes across columns
// (float4 coalesced loads). Computes k_i = row·key_w + key_b and accumulates
// s_partial[c] += k_i * row[c] into per-wave register accumulators, reduced
// deterministically through LDS into a per-block partial vector.
// ---------------------------------------------------------------------------
__global__ __launch_bounds__(256)
void passA_kernel(const float* __restrict__ in,
                  const float* __restrict__ key_w,
                  const float* __restrict__ key_b,
                  float* __restrict__ s_part) {
    __shared__ float acc_sh[8][DIM];            // 16 KB of the 320 KB LDS
    const int lane = threadIdx.x & 31;
    const int wave = threadIdx.x >> 5;
    const float kb = key_b[0];

    float4 kw[4], acc[4];
#pragma unroll
    for (int j = 0; j < 4; ++j) {
        kw[j]  = *(const float4*)(key_w + 4 * lane + 128 * j);
        acc[j] = make_float4(0.f, 0.f, 0.f, 0.f);
    }

    const int row0 = blockIdx.x * ROWS_A;
    for (int r = wave; r < ROWS_A; r += 8) {
        const float* rp = in + (size_t)(row0 + r) * DIM;
        // WGP-scope prefetch (locality 3): fill all cache levels for the next
        // row this wave will consume.
        __builtin_prefetch(rp + 8 * DIM + 4 * lane, 0, 3);
        float4 v[4];
        float dot = 0.f;
#pragma unroll
        for (int j = 0; j < 4; ++j) {
            v[j] = *(const float4*)(rp + 4 * lane + 128 * j); // global_load_b128
            dot += v[j].x * kw[j].x + v[j].y * kw[j].y +
                   v[j].z * kw[j].z + v[j].w * kw[j].w;
        }
#pragma unroll
        for (int off = 16; off > 0; off >>= 1)               // wave32 reduction
            dot += __shfl_xor(dot, off, 32);
        const float kv = dot + kb;
#pragma unroll
        for (int j = 0; j < 4; ++j) {
            acc[j].x += kv * v[j].x;  acc[j].y += kv * v[j].y;
            acc[j].z += kv * v[j].z;  acc[j].w += kv * v[j].w;
        }
    }
#pragma unroll
    for (int j = 0; j < 4; ++j) {
        const int c = 4 * lane + 128 * j;
        acc_sh[wave][c + 0] = acc[j].x;  acc_sh[wave][c + 1] = acc[j].y;
        acc_sh[wave][c + 2] = acc[j].z;  acc_sh[wave][c + 3] = acc[j].w;
    }
    __syncthreads();
    for (int c = threadIdx.x; c < DIM; c += 256) {
        float s = 0.f;
#pragma unroll
        for (int w = 0; w < 8; ++w) s += acc_sh[w][c];
        s_part[(size_t)blockIdx.x * DIM + c] = s;            // deterministic partial
    }
}

// ---------------------------------------------------------------------------
// Reduce s partials, then t[c] = sum_r value_w[r][c]*s[r]  (coalesced column
// reads), and cbias = value_b·s. One block of 512 threads; ~3 MB traffic.
// ---------------------------------------------------------------------------
__global__ __launch_bounds__(512)
void reduceS_T_kernel(const float* __restrict__ s_part,
                      const float* __restrict__ value_w,
                      const float* __restrict__ value_b,
                      float* __restrict__ t_out,
                      float* __restrict__ c_out) {
    __shared__ float s_sh[DIM];
    __shared__ float red[DIM];
    const int d = threadIdx.x;

    float s = 0.f;
    for (int p = 0; p < GRID_A; ++p) s += s_part[(size_t)p * DIM + d];
    s_sh[d] = s;
    __syncthreads();

    float tv = 0.f;
    for (int r = 0; r < DIM; ++r) tv += value_w[(size_t)r * DIM + d] * s_sh[r];
    t_out[d] = tv;

    red[d] = value_b[d] * s_sh[d];
    __syncthreads();
    for (int st = DIM / 2; st > 0; st >>= 1) {
        if (d < st) red[d] += red[d + st];
        __syncthreads();
    }
    if (d == 0) c_out[0] = red[0];
}

// ---------------------------------------------------------------------------
// Pass B: x_i = inputs_i·t + c via V_WMMA_F32_16X16X4_F32. One wave = 16 rows
// as the 16x4 f32 A tile (lane m=lane&15 holds row m; lanes>=16 hold K=2,3).
// t occupies column 0 of B (masked arithmetically -> no EXEC divergence around
// the WMMA). Column 0 of D lands in lanes 0 (M=0..7) and 16 (M=8..15).
// Also emits per-block online-softmax partials (max, sum-exp).
// ---------------------------------------------------------------------------
__global__ __launch_bounds__(256)
void passB_kernel(const float* __restrict__ in,
                  const float* __restrict__ t_in,
                  const float* __restrict__ c_in,
                  float* __restrict__ x_out,
                  float* __restrict__ m_part,
                  float* __restrict__ e_part) {
    __shared__ float t_sh[DIM];
    __shared__ float x_sh[ROWS_B];
    __shared__ float red[256];

    for (int i = threadIdx.x; i < DIM; i += 256) t_sh[i] = t_in[i];
    const float cb = c_in[0];
    __syncthreads();

    const int lane  = threadIdx.x & 31;
    const int wave  = threadIdx.x >> 5;
    const int m     = lane & 15;
    const int khalf = (lane >> 4) << 1;                 // 0 or 2
    const float bmask = (m == 0) ? 1.0f : 0.0f;         // column-0 select, branch-free
    const int row0 = blockIdx.x * ROWS_B + wave * 16;
    const float* rp = in + (size_t)(row0 + m) * DIM + khalf;

    v8f acc = {};
    for (int k = 0; k < DIM; k += 32) {
        // WGP-scope prefetch (locality 3): the demand loads below are
        // 8B-granular, so the 128B line must live in the near cache.
        __builtin_prefetch(rp + k + 128, 0, 3);         // 512B ahead
#pragma unroll
        for (int kk = 0; kk < 32; kk += 4) {
            const float2 a2 = *(const float2*)(rp + k + kk);  // global_load_b64
            v2f a; a.x = a2.x; a.y = a2.y;                    // A[m][k..k+1] / [k+2..k+3]
            v2f b;
            b.x = t_sh[k + kk + khalf]     * bmask;           // B[K][0] = t[K]
            b.y = t_sh[k + kk + khalf + 1] * bmask;
            acc = __builtin_amdgcn_wmma_f32_16x16x4_f32(
                false, a, false, b, (short)0, acc, false, false);
        }
    }

    if (m == 0) {                                       // lanes 0 and 16 hold column 0
        const int rb = wave * 16 + (lane >> 4) * 8;     // rows rb..rb+7 of this block
#pragma unroll
        for (int j = 0; j < 8; ++j) {
            const float xv = acc[j] + cb;
            x_out[(size_t)blockIdx.x * ROWS_B + rb + j] = xv;
            x_sh[rb + j] = xv;
        }
    }
    __syncthreads();

    // Block-local online-softmax partials over the 128 x values.
    red[threadIdx.x] = (threadIdx.x < ROWS_B) ? x_sh[threadIdx.x] : -3.402823466e38f;
    __syncthreads();
    for (int s = 128; s > 0; s >>= 1) {
        if (threadIdx.x < s)
            red[threadIdx.x] = fmaxf(red[threadIdx.x], red[threadIdx.x + s]);
        __syncthreads();
    }
    const float mb = red[0];
    __syncthreads();
    red[threadIdx.x] = (threadIdx.x < ROWS_B) ? __expf(x_sh[threadIdx.x] - mb) : 0.f;
    __syncthreads();
    for (int s = 128; s > 0; s >>= 1) {
        if (threadIdx.x < s) red[threadIdx.x] += red[threadIdx.x + s];
        __syncthreads();
    }
    if (threadIdx.x == 0) { m_part[blockIdx.x] = mb; e_part[blockIdx.x] = red[0]; }
}

// ---------------------------------------------------------------------------
// Combine the 512 (max, sum-exp) partials: M = max m_b; S = sum s_b*exp(m_b-M).
// ---------------------------------------------------------------------------
__global__ __launch_bounds__(512)
void combine_kernel(const float* __restrict__ m_part,
                    const float* __restrict__ e_part,
                    float* __restrict__ fin) {
    __shared__ float red[GRID_B];
    const int i = threadIdx.x;
    red[i] = m_part[i];
    __syncthreads();
    for (int s = GRID_B / 2; s > 0; s >>= 1) {
        if (i < s) red[i] = fmaxf(red[i], red[i + s]);
        __syncthreads();
    }
    const float M = red[0];
    __syncthreads();
    red[i] = e_part[i] * __expf(m_part[i] - M);
    __syncthreads();
    for (int s = GRID_B / 2; s > 0; s >>= 1) {
        if (i < s) red[i] += red[i + s];
        __syncthreads();
    }
    if (i == 0) { fin[0] = M; fin[1] = red[0]; }
}

__global__ __launch_bounds__(256)
void finalize_kernel(const float* __restrict__ x,
                     const float* __restrict__ fin,
                     float* __restrict__ out) {
    const int i = blockIdx.x * 256 + threadIdx.x;
    const float M = fin[0];
    const float invS = 1.0f / fin[1];
    out[i] = __expf(x[i] - M) * invS;
}

// ---------------------------------------------------------------------------
extern "C" void kernel_launch(void* const* d_in, const int* in_sizes, int n_in,
                              void* d_out, int out_size, void* d_ws, size_t ws_size,
                              hipStream_t stream) {
    const float* inputs  = (const float*)d_in[0];   // [N, D]
    const float* key_w   = (const float*)d_in[1];   // [1, D]
    const float* key_b   = (const float*)d_in[2];   // [1]
    const float* value_w = (const float*)d_in[3];   // [D, D]
    const float* value_b = (const float*)d_in[4];   // [D]
    float* out = (float*)d_out;                     // [N]

    // Workspace layout (floats); every region is fully written before read.
    float* ws     = (float*)d_ws;
    float* s_part = ws;                                   // GRID_A * DIM  (2 MB)
    float* t      = s_part + (size_t)GRID_A * DIM;        // DIM
    float* cbias  = t + DIM;                              // 1
    float* x      = cbias + 1;                            // N
    float* m_part = x + N_ROWS;                           // GRID_B
    float* e_part = m_part + GRID_B;                      // GRID_B
    float* fin    = e_part + GRID_B;                      // 2

    passA_kernel   <<<GRID_A, 256, 0, stream>>>(inputs, key_w, key_b, s_part);
    reduceS_T_kernel<<<1, 512, 0, stream>>>(s_part, value_w, value_b, t, cbias);
    passB_kernel   <<<GRID_B, 256, 0, stream>>>(inputs, t, cbias, x, m_part, e_part);
    combine_kernel <<<1, GRID_B, 0, stream>>>(m_part, e_part, fin);
    finalize_kernel<<<N_ROWS / 256, 256, 0, stream>>>(x, fin, out);
}